// CVNeuralNetwork_60610578482013
// MI455X (gfx1250) — compile-verified
//
#include <hip/hip_runtime.h>

// ---------------------------------------------------------------------------
// CV quantum neural network, B=4096 samples, M=4 modes, L=4 layers, D=8.
// All gates canonicalized to 64x64 complex matrices applied on a mode pair.
// State (4096 complex amplitudes / sample) lives in LDS; contraction done
// with V_WMMA_F32_16X16X4_F32 (wave32, fp32 matrix core path on gfx1250).
// Gate matrices are double-buffered in LDS and streamed from L2 with
// GLOBAL_LOAD_ASYNC_TO_LDS_B128 (ASYNCcnt) overlapping the WMMA compute.
// ---------------------------------------------------------------------------

typedef __attribute__((ext_vector_type(2))) float v2f;
typedef __attribute__((ext_vector_type(8))) float v8f;
typedef __attribute__((ext_vector_type(4))) int   v4i;

#define WMMA_F32(a, b, c) \
  __builtin_amdgcn_wmma_f32_16x16x4_f32(false, (a), false, (b), (short)0, (c), false, false)

#define NB 4096      // batch
#define NM 4         // modes
#define NL 4         // layers
#define ND 8         // cutoff
#define NST 4096     // 8^4 amplitudes per sample

#define AS1 __attribute__((address_space(1)))
#define AS3 __attribute__((address_space(3)))

#if __has_builtin(__builtin_amdgcn_global_load_async_to_lds_b128)
#define HAVE_ASYNC_LDS 1
#else
#define HAVE_ASYNC_LDS 0
#endif

#if HAVE_ASYNC_LDS
__device__ inline void async_lds_b128(const float* gsrc, float* ldst) {
  __builtin_amdgcn_global_load_async_to_lds_b128(
      (AS1 v4i*)gsrc, (AS3 v4i*)ldst, 0, 0);
}
__device__ inline void wait_async_zero() {
#if __has_builtin(__builtin_amdgcn_s_wait_asynccnt)
  __builtin_amdgcn_s_wait_asynccnt(0);
#else
  asm volatile("s_wait_asynccnt 0x0" ::: "memory");
#endif
}
#endif

// ------------------------- per-thread 8x8 complex expm ---------------------
// scaling (1/64) + Taylor(10) + 6 squarings; matrices are tiny & well scaled.
__device__ inline void expm8c(const float* Hr, const float* Hi, float* Rr, float* Ri) {
  float tr[64], ti[64], pr[64], pi[64];
  const float sc = 1.0f / 64.0f;
  for (int e = 0; e < 64; ++e) {
    float hr = Hr[e] * sc, hi = Hi[e] * sc;
    tr[e] = hr; ti[e] = hi;
    Rr[e] = hr + (((e >> 3) == (e & 7)) ? 1.0f : 0.0f);
    Ri[e] = hi;
  }
  for (int n = 2; n <= 10; ++n) {
    float inv = 1.0f / (float)n;
    for (int i = 0; i < 8; ++i)
      for (int j = 0; j < 8; ++j) {
        float sr = 0.f, si = 0.f;
        for (int k = 0; k < 8; ++k) {
          float ar = tr[i * 8 + k], ai = ti[i * 8 + k];
          float br = Hr[k * 8 + j] * sc, bi = Hi[k * 8 + j] * sc;
          sr += ar * br - ai * bi;
          si += ar * bi + ai * br;
        }
        pr[i * 8 + j] = sr * inv; pi[i * 8 + j] = si * inv;
      }
    for (int e = 0; e < 64; ++e) {
      tr[e] = pr[e]; ti[e] = pi[e];
      Rr[e] += tr[e]; Ri[e] += ti[e];
    }
  }
  for (int s = 0; s < 6; ++s) {
    for (int i = 0; i < 8; ++i)
      for (int j = 0; j < 8; ++j) {
        float sr = 0.f, si = 0.f;
        for (int k = 0; k < 8; ++k) {
          float ar = Rr[i * 8 + k], ai = Ri[i * 8 + k];
          float br = Rr[k * 8 + j], bi = Ri[k * 8 + j];
          sr += ar * br - ai * bi;
          si += ar * bi + ai * br;
        }
        pr[i * 8 + j] = sr; pi[i * 8 + j] = si;
      }
    for (int e = 0; e < 64; ++e) { Rr[e] = pr[e]; Ri[e] = pi[e]; }
  }
}

// ----------------- kernel 1: per-sample encode displacement expm -----------
// D(x) = expm(x (AD - A)) real 8x8, one thread per (b, mode).
__global__ __launch_bounds__(256) void encode_dx(const float* __restrict__ x,
                                                 float* __restrict__ WDX) {
  int id = blockIdx.x * blockDim.x + threadIdx.x;
  if (id >= NB * NM) return;
  float xv = x[id];
  float Hr[64], Hi[64], Ur[64], Ui[64];
  for (int e = 0; e < 64; ++e) { Hr[e] = 0.f; Hi[e] = 0.f; }
  for (int k = 0; k < 7; ++k) Hr[(k + 1) * 8 + k] += xv * sqrtf((float)(k + 1)); // AD term
  for (int k = 1; k < 8; ++k) Hr[(k - 1) * 8 + k] -= xv * sqrtf((float)k);       // -A term
  expm8c(Hr, Hi, Ur, Ui);
  for (int e = 0; e < 64; ++e) WDX[(size_t)id * 64 + e] = Ur[e];
}

// ------------------- LDS-cooperative 64x64 complex expm --------------------
__device__ inline void cmm64(const float* ar, const float* ai,
                             const float* br, const float* bi,
                             float* cr, float* ci, int tid) {
  for (int idx = tid; idx < 4096; idx += 256) {
    int i = idx >> 6, j = idx & 63;
    float sr = 0.f, si = 0.f;
    for (int k = 0; k < 64; ++k) {
      float xr = ar[i * 64 + k], xi = ai[i * 64 + k];
      float yr = br[k * 64 + j], yi = bi[k * 64 + j];
      sr += xr * yr - xi * yi;
      si += xr * yi + xi * yr;
    }
    cr[idx] = sr; ci[idx] = si;
  }
}

__device__ inline void expm64c(float* HR, float* HI, float* TR, float* TI,
                               float* RR, float* RI, float* PR, float* PI, int tid) {
  const float sc = 1.0f / 1024.0f;   // 2^-10 scaling
  for (int idx = tid; idx < 4096; idx += 256) {
    HR[idx] *= sc; HI[idx] *= sc;
    float d = ((idx >> 6) == (idx & 63)) ? 1.0f : 0.0f;
    TR[idx] = HR[idx]; TI[idx] = HI[idx];
    RR[idx] = HR[idx] + d; RI[idx] = HI[idx];
  }
  __syncthreads();
  for (int n = 2; n <= 8; ++n) {
    cmm64(TR, TI, HR, HI, PR, PI, tid);
    __syncthreads();
    float inv = 1.0f / (float)n;
    for (int idx = tid; idx < 4096; idx += 256) {
      TR[idx] = PR[idx] * inv; TI[idx] = PI[idx] * inv;
      RR[idx] += TR[idx];      RI[idx] += TI[idx];
    }
    __syncthreads();
  }
  for (int s = 0; s < 10; ++s) {
    cmm64(RR, RI, RR, RI, PR, PI, tid);
    __syncthreads();
    for (int idx = tid; idx < 4096; idx += 256) { RR[idx] = PR[idx]; RI[idx] = PI[idx]; }
    __syncthreads();
  }
}

// squeeze H = 0.5(conj(z) A^2 - z AD^2), z = r e^{i phi}
__device__ inline void build_sq_H(float r, float ph, float* Hr, float* Hi) {
  for (int e = 0; e < 64; ++e) { Hr[e] = 0.f; Hi[e] = 0.f; }
  float zr = r * cosf(ph), zi = r * sinf(ph);
  for (int k = 2; k < 8; ++k) {
    float s = 0.5f * sqrtf((float)(k * (k - 1)));
    Hr[(k - 2) * 8 + k] += zr * s; Hi[(k - 2) * 8 + k] -= zi * s;
  }
  for (int k = 0; k < 6; ++k) {
    float s = 0.5f * sqrtf((float)((k + 1) * (k + 2)));
    Hr[(k + 2) * 8 + k] -= zr * s; Hi[(k + 2) * 8 + k] -= zi * s;
  }
}

// displacement H = alpha AD - conj(alpha) A, alpha = (r cos ph) e^{i r sin ph}
__device__ inline void build_disp_H(float r, float ph, float* Hr, float* Hi) {
  for (int e = 0; e < 64; ++e) { Hr[e] = 0.f; Hi[e] = 0.f; }
  float m = r * cosf(ph), th = r * sinf(ph);
  float alr = m * cosf(th), ali = m * sinf(th);
  for (int k = 0; k < 7; ++k) {
    float s = sqrtf((float)(k + 1));
    Hr[(k + 1) * 8 + k] += alr * s; Hi[(k + 1) * 8 + k] += ali * s;
  }
  for (int k = 1; k < 8; ++k) {
    float s = sqrtf((float)k);
    Hr[(k - 1) * 8 + k] -= alr * s; Hi[(k - 1) * 8 + k] += ali * s;
  }
}

// ----------------- kernel 2: build the 64 shared 64x64 gate matrices -------
// slot t in layer l:  t0..5  interferometer-1 beamsplitters (rot diags folded)
//                     t6..7  squeeze krons on (0,1) / (2,3)
//                     t8..13 interferometer-2 beamsplitters
//                     t14,15 displacement krons with Kerr diag folded
// Stored TRANSPOSED (input-major): WG[g][c*64 + n] so it stages directly as
// the WMMA B operand (K rows = input index c, N cols = output index n).
__global__ __launch_bounds__(256) void build_gates(
    const float* __restrict__ th1, const float* __restrict__ ph1,
    const float* __restrict__ th2, const float* __restrict__ ph2,
    const float* __restrict__ dr,  const float* __restrict__ dphi,
    const float* __restrict__ sr,  const float* __restrict__ sphi,
    const float* __restrict__ kerr, float* __restrict__ WG) {
  extern __shared__ float sm[];
  float* HR = sm;          float* HI = sm + 4096;
  float* TR = sm + 8192;   float* TI = sm + 12288;
  float* RR = sm + 16384;  float* RI = sm + 20480;
  float* PR = sm + 24576;  float* PIm = sm + 28672;
  const int g = blockIdx.x, l = g >> 4, t = g & 15, tid = threadIdx.x;
  float* out = WG + (size_t)g * 8192;
  const int PMi[6] = {0, 0, 0, 1, 1, 2};
  const int PMj[6] = {1, 2, 3, 2, 3, 3};

  if (t < 6 || (t >= 8 && t < 14)) {
    // ---- beamsplitter with folded rotation diagonals ----
    int pr = (t < 6) ? t : t - 8;
    const float* TH = (t < 6) ? th1 : th2;
    const float* PH = (t < 6) ? ph1 : ph2;
    int i = PMi[pr], j = PMj[pr];
    float theta = TH[l * 16 + i * 4 + j];
    float phi   = PH[l * 16 + i * 4 + j];
    float cp = cosf(phi), sp = sinf(phi);
    for (int idx = tid; idx < 4096; idx += 256) {
      int n = idx >> 6, c = idx & 63;
      int a = n >> 3, b = n & 7, k = c >> 3, lq = c & 7;
      float hr = 0.f, hi = 0.f;
      if (a == k - 1 && b == lq + 1) {           //  theta e^{i phi} a1 a2^dag
        float m = theta * sqrtf((float)k) * sqrtf((float)(lq + 1));
        hr += m * cp; hi += m * sp;
      }
      if (a == k + 1 && b == lq - 1) {           // -theta e^{-i phi} a1^dag a2
        float m = theta * sqrtf((float)(k + 1)) * sqrtf((float)lq);
        hr -= m * cp; hi += m * sp;
      }
      HR[n * 64 + c] = hr; HI[n * 64 + c] = hi;
    }
    __syncthreads();
    expm64c(HR, HI, TR, TI, RR, RI, PR, PIm, tid);
    for (int idx = tid; idx < 4096; idx += 256) {
      int n = idx >> 6, c = idx & 63;
      int b = n & 7, k = c >> 3, lq = c & 7;
      // pre-rotations folded onto the input index (first BS touching a mode)
      float prep = 0.f;
      if (pr == 0)      prep = PH[l * 16 + 0]  * (float)k + PH[l * 16 + 5] * (float)lq;
      else if (pr == 1) prep = PH[l * 16 + 10] * (float)lq;
      else if (pr == 2) prep = PH[l * 16 + 15] * (float)lq;
      // post-rotation phi[j,i] on mode j folded onto the output index
      float postp = PH[l * 16 + j * 4 + i] * (float)b;
      float ang = prep + postp, ca = cosf(ang), sa = sinf(ang);
      float ur = RR[n * 64 + c], ui = RI[n * 64 + c];
      out[c * 64 + n]        = ur * ca - ui * sa;
      out[4096 + c * 64 + n] = ur * sa + ui * ca;
    }
  } else if (t == 6 || t == 7) {
    // ---- squeeze kron: S_i (x) S_j on pair (0,1) or (2,3) ----
    int pq = (t == 6) ? 0 : 2, wi = pq, wj = pq + 1;
    float H1r[64], H1i[64], U1r[64], U1i[64];
    float H2r[64], H2i[64], U2r[64], U2i[64];
    build_sq_H(sr[l * 4 + wi], sphi[l * 4 + wi], H1r, H1i);
    build_sq_H(sr[l * 4 + wj], sphi[l * 4 + wj], H2r, H2i);
    expm8c(H1r, H1i, U1r, U1i);
    expm8c(H2r, H2i, U2r, U2i);
    for (int idx = tid; idx < 4096; idx += 256) {
      int n = idx >> 6, c = idx & 63;
      int a = n >> 3, b = n & 7, k = c >> 3, lq = c & 7;
      float xr = U1r[a * 8 + k], xi = U1i[a * 8 + k];
      float yr = U2r[b * 8 + lq], yi = U2i[b * 8 + lq];
      out[c * 64 + n]        = xr * yr - xi * yi;
      out[4096 + c * 64 + n] = xr * yi + xi * yr;
    }
  } else {
    // ---- displacement kron with Kerr diag folded on output ----
    int pq = (t == 14) ? 0 : 2, wi = pq, wj = pq + 1;
    float H1r[64], H1i[64], U1r[64], U1i[64];
    float H2r[64], H2i[64], U2r[64], U2i[64];
    build_disp_H(dr[l * 4 + wi], dphi[l * 4 + wi], H1r, H1i);
    build_disp_H(dr[l * 4 + wj], dphi[l * 4 + wj], H2r, H2i);
    expm8c(H1r, H1i, U1r, U1i);
    expm8c(H2r, H2i, U2r, U2i);
    float ki = kerr[l * 4 + wi], kj = kerr[l * 4 + wj];
    for (int idx = tid; idx < 4096; idx += 256) {
      int n = idx >> 6, c = idx & 63;
      int a = n >> 3, b = n & 7, k = c >> 3, lq = c & 7;
      float xr = U1r[a * 8 + k], xi = U1i[a * 8 + k];
      float yr = U2r[b * 8 + lq], yi = U2i[b * 8 + lq];
      float ur = xr * yr - xi * yi, ui = xr * yi + xi * yr;
      float ang = ki * (float)(a * a) + kj * (float)(b * b);  // Kerr: e^{i kappa n^2}
      float ca = cosf(ang), sa = sinf(ang);
      out[c * 64 + n]        = ur * ca - ui * sa;
      out[4096 + c * 64 + n] = ur * sa + ui * ca;
    }
  }
}

// -------------- kernel 3: state evolution, one sample per block ------------
// LDS: ping-pong complex state (4 x 4096 f32) + double-buffered gate
// (2 x 2 x 4096 f32). 8 waves: wave w owns row-tile rt=w/2 and column tiles
// {2(w&1), 2(w&1)+1}. Complex GEMM = 4 real WMMA chains: 8 wmma per k-step.
// Gate for slot s+1 streams L2 -> LDS asynchronously during slot s compute.
__global__ __launch_bounds__(256) void evolve(const float* __restrict__ WDX,
                                              const float* __restrict__ WG,
                                              float* __restrict__ out) {
  extern __shared__ float sm[];
  float* S0r = sm;            float* S0i = sm + 4096;
  float* S1r = sm + 8192;     float* S1i = sm + 12288;
  float* Br0 = sm + 16384;    float* Bi0 = sm + 20480;
  float* Br1 = sm + 24576;    float* Bi1 = sm + 28672;
  float* bins = sm + 32768;   // 32 floats: 4 modes x 8 levels
  const int b = blockIdx.x;
  const int tid = threadIdx.x, lane = tid & 31, w = tid >> 5;
  const int rt = w >> 1, cg = w & 1;
  const int hk = (lane >> 4) << 1;     // lane<16 -> K {0,1}, lane>=16 -> K {2,3}
  const int ncol = lane & 15;
  const int n0 = cg * 32 + ncol, n1 = n0 + 16;

  for (int idx = tid; idx < 4096; idx += 256) {
    S0r[idx] = (idx == 0) ? 1.0f : 0.0f;  S0i[idx] = 0.0f;   // vacuum
    S1r[idx] = 0.0f;                      S1i[idx] = 0.0f;
  }
  __syncthreads();

  const int PMi[6] = {0, 0, 0, 1, 1, 2};
  const int PMj[6] = {1, 2, 3, 2, 3, 3};
  int p = 0;

  for (int slot = 0; slot < 66; ++slot) {
    int pairid; int encp = -1;
    if (slot == 0)      { pairid = 0; encp = 0; }
    else if (slot == 1) { pairid = 5; encp = 2; }
    else {
      int t = (slot - 2) & 15;
      if (t < 6) pairid = t;
      else if (t == 6) pairid = 0;
      else if (t == 7) pairid = 5;
      else if (t < 14) pairid = t - 8;
      else if (t == 14) pairid = 0;
      else pairid = 5;
    }
    const int cur = slot & 1;
    float* Bcr = cur ? Br1 : Br0;
    float* Bci = cur ? Bi1 : Bi0;

    // ---- stage current B operand (encode slots build the kron in-place) ----
    if (encp >= 0) {
      const float* D0 = WDX + ((size_t)b * 4 + encp) * 64;
      const float* D1 = WDX + ((size_t)b * 4 + encp + 1) * 64;
      for (int idx = tid; idx < 4096; idx += 256) {
        int c = idx >> 6, n = idx & 63;
        int k = c >> 3, lq = c & 7, i = n >> 3, jq = n & 7;
        Bcr[idx] = D0[i * 8 + k] * D1[jq * 8 + lq];   // real kron, Bi = 0
        Bci[idx] = 0.0f;
      }
    }
#if !HAVE_ASYNC_LDS
    else {
      const float* gptr = WG + (size_t)(slot - 2) * 8192;
      const float4* src = (const float4*)gptr;
      float4* dR = (float4*)Bcr; float4* dI = (float4*)Bci;
      for (int q = tid; q < 1024; q += 256) { dR[q] = src[q]; dI[q] = src[q + 1024]; }
      if (slot < 65) __builtin_prefetch(gptr + 8192, 0, 0);  // next gate -> L2
    }
#else
    // ---- kick off async L2->LDS stream of the NEXT gate matrix ----
    if (slot >= 1 && slot < 65) {
      const float* gn = WG + (size_t)(slot - 1) * 8192;   // gate for slot+1
      float* Bnr = cur ? Br0 : Br1;
      float* Bni = cur ? Bi0 : Bi1;
      for (int r = 0; r < 8; ++r) {
        int q = tid + r * 256;                            // 2048 b128 chunks
        float* ldst = (q < 1024) ? (Bnr + q * 4) : (Bni + (q - 1024) * 4);
        async_lds_b128(gn + q * 4, ldst);
      }
    }
#endif
    __syncthreads();

    const int m1 = PMi[pairid], m2 = PMj[pairid];
    const int sm1 = 512 >> (3 * m1), sm2 = 512 >> (3 * m2);
    int o1 = -1, o2 = -1;
    for (int m = 0; m < 4; ++m)
      if (m != m1 && m != m2) { if (o1 < 0) o1 = m; else o2 = m; }
    const int so1 = 512 >> (3 * o1), so2 = 512 >> (3 * o2);

    const float* Sr = p ? S1r : S0r; const float* Si = p ? S1i : S0i;
    float* Dr = p ? S0r : S1r;       float* Di = p ? S0i : S1i;

    const int arow = rt * 16 + (lane & 15);
    const int rowbase = (arow >> 3) * so1 + (arow & 7) * so2;

    v8f cr0 = {0, 0, 0, 0, 0, 0, 0, 0};
    v8f ci0 = cr0, cr1 = cr0, ci1 = cr0;

#pragma unroll
    for (int kk = 0; kk < 16; ++kk) {
      int c0 = kk * 4 + hk, c1 = c0 + 1;
      int off0 = rowbase + (c0 >> 3) * sm1 + (c0 & 7) * sm2;
      int off1 = rowbase + (c1 >> 3) * sm1 + (c1 & 7) * sm2;
      v2f ar, ai;
      ar.x = Sr[off0]; ar.y = Sr[off1];
      ai.x = Si[off0]; ai.y = Si[off1];
      int bb0 = c0 * 64, bb1 = c1 * 64;
      v2f br0, bi0, br1, bi1;
      br0.x = Bcr[bb0 + n0]; br0.y = Bcr[bb1 + n0];
      bi0.x = Bci[bb0 + n0]; bi0.y = Bci[bb1 + n0];
      br1.x = Bcr[bb0 + n1]; br1.y = Bcr[bb1 + n1];
      bi1.x = Bci[bb0 + n1]; bi1.y = Bci[bb1 + n1];
      v2f bin0 = -bi0, bin1 = -bi1;
      cr0 = WMMA_F32(ar, br0, cr0);
      cr0 = WMMA_F32(ai, bin0, cr0);
      ci0 = WMMA_F32(ar, bi0, ci0);
      ci0 = WMMA_F32(ai, br0, ci0);
      cr1 = WMMA_F32(ar, br1, cr1);
      cr1 = WMMA_F32(ai, bin1, cr1);
      ci1 = WMMA_F32(ar, bi1, ci1);
      ci1 = WMMA_F32(ai, br1, ci1);
    }

    // writeback to the other ping-pong buffer (no hazard with readers)
#pragma unroll
    for (int v = 0; v < 8; ++v) {
      int row = rt * 16 + v + ((lane >> 4) << 3);
      int rb = (row >> 3) * so1 + (row & 7) * so2;
      int d0 = rb + (n0 >> 3) * sm1 + (n0 & 7) * sm2;
      int d1 = rb + (n1 >> 3) * sm1 + (n1 & 7) * sm2;
      Dr[d0] = cr0[v]; Di[d0] = ci0[v];
      Dr[d1] = cr1[v]; Di[d1] = ci1[v];
    }
#if HAVE_ASYNC_LDS
    wait_async_zero();   // next gate fully landed in LDS
#endif
    __syncthreads();
    p ^= 1;
  }

  // ---- <n_w> per mode: marginal |psi|^2 via ds_add_f32 bins ----
  if (tid < 32) bins[tid] = 0.0f;
  __syncthreads();
  const float* Fr = p ? S1r : S0r; const float* Fi = p ? S1i : S0i;
  for (int idx = tid; idx < 4096; idx += 256) {
    float prb = Fr[idx] * Fr[idx] + Fi[idx] * Fi[idx];
    atomicAdd(&bins[0 * 8 + ((idx >> 9) & 7)], prb);
    atomicAdd(&bins[1 * 8 + ((idx >> 6) & 7)], prb);
    atomicAdd(&bins[2 * 8 + ((idx >> 3) & 7)], prb);
    atomicAdd(&bins[3 * 8 + (idx & 7)], prb);
  }
  __syncthreads();
  if (tid < 4) {
    float ev = 0.0f;
    for (int n = 0; n < 8; ++n) ev += (float)n * bins[tid * 8 + n];
    out[(size_t)b * 4 + tid] = ev;
  }
}

// ---------------------------------------------------------------------------
extern "C" void kernel_launch(void* const* d_in, const int* in_sizes, int n_in,
                              void* d_out, int out_size, void* d_ws, size_t ws_size,
                              hipStream_t stream) {
  const float* x    = (const float*)d_in[0];
  const float* th1  = (const float*)d_in[1];
  const float* ph1  = (const float*)d_in[2];
  const float* th2  = (const float*)d_in[3];
  const float* ph2  = (const float*)d_in[4];
  const float* dr   = (const float*)d_in[5];
  const float* dphi = (const float*)d_in[6];
  const float* sqr  = (const float*)d_in[7];
  const float* sqph = (const float*)d_in[8];
  const float* kerr = (const float*)d_in[9];
  float* out = (float*)d_out;

  float* WDX = (float*)d_ws;               // B*M*64   = 1,048,576 f32 (4 MB)
  float* WG  = WDX + (size_t)NB * NM * 64; // 64*8192  =   524,288 f32 (2 MB)

  encode_dx<<<(NB * NM + 255) / 256, 256, 0, stream>>>(x, WDX);
  build_gates<<<64, 256, 8 * 4096 * sizeof(float), stream>>>(
      th1, ph1, th2, ph2, dr, dphi, sqr, sqph, kerr, WG);
  evolve<<<NB, 256, (8 * 4096 + 32) * sizeof(float), stream>>>(WDX, WG, out);
}